// GCNLinkPredictor_68195490725940
// MI455X (gfx1250) — compile-verified
//
#include <hip/hip_runtime.h>
#include <hip/hip_bf16.h>
#include <math.h>

typedef __attribute__((ext_vector_type(2))) float v2f;
typedef __attribute__((ext_vector_type(8))) float v8f;
typedef __attribute__((ext_vector_type(4))) unsigned int u32x4;
typedef __attribute__((ext_vector_type(8))) int i32x8;
typedef __attribute__((ext_vector_type(4))) int i32x4;

#define FEAT 128

// ---------------- degree / normalization ----------------
__global__ void deg_init_kernel(float* __restrict__ deg, int n) {
    int i = blockIdx.x * blockDim.x + threadIdx.x;
    if (i < n) deg[i] = 1.0f;   // self loop
}

__global__ void deg_edge_kernel(const long long* __restrict__ dst,
                                float* __restrict__ deg, int e) {
    int i = blockIdx.x * blockDim.x + threadIdx.x;
    if (i < e) atomicAdd(&deg[(int)dst[i]], 1.0f);
}

__global__ void dinv_kernel(float* __restrict__ deg, int n) {
    int i = blockIdx.x * blockDim.x + threadIdx.x;
    if (i < n) deg[i] = rsqrtf(deg[i]);
}

__global__ void zero4_kernel(float4* __restrict__ p, long long n4) {
    long long i = (long long)blockIdx.x * blockDim.x + threadIdx.x;
    if (i < n4) p[i] = make_float4(0.f, 0.f, 0.f, 0.f);
}

// ---------------- Tensor Data Mover: 2D tile -> LDS ----------------
// Builds a D# per CDNA5 ISA ch.8 and issues TENSOR_LOAD_TO_LDS.
// tile_d0   : row length in f32 elements (tile_dim0 / tensor_dim0)
// tile_d1   : number of rows (tile_dim1)
// tensor_d1 : rows actually in-bounds (OOB rows read as zero)
// stride0   : row stride in f32 elements (tensor_dim0_stride)
// pad codes : LDS padding (interval code: 5 -> 64 dwords, amount code: 63 -> 64 dwords)
__device__ __forceinline__ void tdm_load_tile(unsigned lds_byte_off,
                                              const float* gsrc,
                                              unsigned tile_d0, unsigned tile_d1,
                                              unsigned tensor_d1, unsigned stride0,
                                              unsigned pad_int_code,
                                              unsigned pad_amt_code,
                                              unsigned pad_enable)
{
    const unsigned long long ga = (unsigned long long)(uintptr_t)gsrc;
    u32x4 g0;
    g0.x = 1u;                                              // count=1, user mode
    g0.y = lds_byte_off;                                    // lds_addr (bytes)
    g0.z = (unsigned)ga;                                    // global_addr[31:0]
    g0.w = (unsigned)((ga >> 32) & 0x01ffffffu) | (2u << 30); // addr[56:32] | type=2
    i32x8 g1;
    g1[0] = (int)((2u << 16) |                              // data_size = 4 bytes
                  ((pad_enable & 1u) << 20) |
                  ((pad_int_code & 7u) << 22) |
                  ((pad_amt_code & 127u) << 25));
    g1[1] = (int)((tile_d0 & 0xffffu) << 16);               // tensor_dim0 lo16
    g1[2] = (int)(((tile_d0 >> 16) & 0xffffu) |             // tensor_dim0 hi16
                  ((tensor_d1 & 0xffffu) << 16));           // tensor_dim1 lo16
    g1[3] = (int)(((tensor_d1 >> 16) & 0xffffu) |           // tensor_dim1 hi16
                  ((tile_d0 & 0xffffu) << 16));             // tile_dim0
    g1[4] = (int)(tile_d1 & 0xffffu);                       // tile_dim1 (tile_dim2=0)
    g1[5] = (int)stride0;                                   // tensor_dim0_stride lo32
    g1[6] = 0;                                              // stride0 hi16 | dim1_stride lo
    g1[7] = 0;
    const i32x4 z4 = {0, 0, 0, 0};
    const i32x8 z8 = {0, 0, 0, 0, 0, 0, 0, 0};
    __builtin_amdgcn_tensor_load_to_lds(g0, g1, z4, z4, z8, 0);
}

// ---------------- WMMA GEMM: g = (X @ W) * dinv[:,None] ----------------
// MODE 0: X = concat(xa [n,64], xb [n,64]) -> [n,128]   MODE 1: X = xa [n,128]
// One block (8 waves) = one 16-row tile; wave w computes the 16-col tile w.
// A tile staged in LDS by the TDM; K=128 via 32x V_WMMA_F32_16X16X4_F32.
template <int MODE>
__global__ __launch_bounds__(256)
void gemm_g_kernel(const float* __restrict__ xa, const float* __restrict__ xb,
                   const float* __restrict__ W, const float* __restrict__ dinv,
                   float* __restrict__ g, int n)
{
    __shared__ float At[16 * FEAT];                 // [16 rows][128 feats]
    const int rt   = blockIdx.x;                    // row tile (16 nodes)
    const int ct   = threadIdx.x >> 5;              // wave id = col tile (16 outputs)
    const int lane = threadIdx.x & 31;

    if (ct == 0) {                                  // wave 0 drives the TDM
        const unsigned lds0 = (unsigned)(uintptr_t)(&At[0]);
        const unsigned rows_left = (unsigned)(n - rt * 16);
        const unsigned td1 = rows_left < 16u ? rows_left : 16u;
        if (MODE == 0) {
            // x_i -> cols 0..63, x_j -> cols 64..127 via 64-dword LDS padding
            tdm_load_tile(lds0,           xa + (size_t)rt * 16 * 64, 64, 16, td1, 64, 5u, 63u, 1u);
            tdm_load_tile(lds0 + 64 * 4,  xb + (size_t)rt * 16 * 64, 64, 16, td1, 64, 5u, 63u, 1u);
        } else {
            tdm_load_tile(lds0, xa + (size_t)rt * 16 * FEAT, FEAT, 16, td1, FEAT, 0u, 0u, 0u);
        }
        __builtin_amdgcn_s_wait_tensorcnt(0);
    }
    __syncthreads();

    const int m     = lane & 15;
    const int khalf = lane >> 4;                    // 0 or 1
    const int col   = ct * 16 + m;
    const float* wp = W + col;

    v8f acc = {};
    #pragma unroll 4
    for (int k = 0; k < FEAT; k += 4) {
        const int ka = k + 2 * khalf;               // this lane: K = ka, ka+1
        const v2f a = *(const v2f*)(&At[m * FEAT + ka]);   // ds_load_b64
        v2f b;
        b.x = wp[(size_t)(ka + 0) * FEAT];
        b.y = wp[(size_t)(ka + 1) * FEAT];
        acc = __builtin_amdgcn_wmma_f32_16x16x4_f32(
                  false, a, false, b, (short)0, acc, false, false);
    }

    // C/D layout: VGPR r holds (M = r + 8*khalf, N = lane&15)
    const int rbase = rt * 16 + khalf * 8;
    if (rt * 16 + 16 <= n) {                        // full tile: no guards
        #pragma unroll
        for (int r = 0; r < 8; ++r)
            g[(size_t)(rbase + r) * FEAT + col] = acc[r] * dinv[rbase + r];
    } else {
        #pragma unroll
        for (int r = 0; r < 8; ++r) {
            const int orow = rbase + r;
            if (orow < n) g[(size_t)orow * FEAT + col] = acc[r] * dinv[orow];
        }
    }
}

// ---------------- edge scatter: acc[dst] += g[src] (128 feats, wave/edge) ----
__global__ __launch_bounds__(256)
void scatter_kernel(const float* __restrict__ g,
                    const long long* __restrict__ src,
                    const long long* __restrict__ dst,
                    float* __restrict__ acc, int e)
{
    const int edge = blockIdx.x * (blockDim.x >> 5) + (threadIdx.x >> 5);
    if (edge >= e) return;
    const int lane = threadIdx.x & 31;
    const int s = (int)src[edge];
    const int d = (int)dst[edge];
    if (lane == 0) {                                 // pull a future source row toward L2
        const int ep = edge + 2048;
        if (ep < e) __builtin_prefetch(g + (size_t)((int)src[ep]) * FEAT, 0, 1);
    }
    const float4 v = *(const float4*)(g + (size_t)s * FEAT + lane * 4);
    float* ap = acc + (size_t)d * FEAT + lane * 4;
    atomicAdd(ap + 0, v.x);
    atomicAdd(ap + 1, v.y);
    atomicAdd(ap + 2, v.z);
    atomicAdd(ap + 3, v.w);
}

// ---------------- epilogue: out = act(dinv*(acc+g) + b), float4-wide -------
__global__ void finalize_kernel(const float4* __restrict__ g,
                                const float4* __restrict__ acc,
                                const float* __restrict__ dinv,
                                const float4* __restrict__ b,
                                float4* __restrict__ out, int n, int do_relu)
{
    long long i = (long long)blockIdx.x * blockDim.x + threadIdx.x;
    if (i >= (long long)n * (FEAT / 4)) return;
    const int v = (int)(i >> 5), f = (int)(i & 31);
    const float di = dinv[v];
    const float4 a = acc[i], gg = g[i], bb = b[f];
    float4 r;
    r.x = di * (a.x + gg.x) + bb.x;
    r.y = di * (a.y + gg.y) + bb.y;
    r.z = di * (a.z + gg.z) + bb.z;
    r.w = di * (a.w + gg.w) + bb.w;
    if (do_relu) {
        r.x = fmaxf(r.x, 0.f); r.y = fmaxf(r.y, 0.f);
        r.z = fmaxf(r.z, 0.f); r.w = fmaxf(r.w, 0.f);
    }
    out[i] = r;
}

// ---------------- layer 3: g3 = (x @ W3) * dinv (wave per node) ------------
__global__ __launch_bounds__(256)
void dot3_kernel(const float* __restrict__ x, const float* __restrict__ W3,
                 const float* __restrict__ dinv, float* __restrict__ g3, int n)
{
    const int node = blockIdx.x * (blockDim.x >> 5) + (threadIdx.x >> 5);
    if (node >= n) return;
    const int lane = threadIdx.x & 31;
    const float4 xv = *(const float4*)(x  + (size_t)node * FEAT + lane * 4);
    const float4 wv = *(const float4*)(W3 + lane * 4);
    float s = xv.x * wv.x + xv.y * wv.y + xv.z * wv.z + xv.w * wv.w;
    #pragma unroll
    for (int off = 16; off > 0; off >>= 1) s += __shfl_down(s, off, 32);
    if (lane == 0) g3[node] = s * dinv[node];
}

__global__ void scatter3_kernel(const float* __restrict__ g3,
                                const long long* __restrict__ src,
                                const long long* __restrict__ dst,
                                float* __restrict__ acc3, int e)
{
    int i = blockIdx.x * blockDim.x + threadIdx.x;
    if (i < e) atomicAdd(&acc3[(int)dst[i]], g3[(int)src[i]]);
}

__global__ void finalize3_kernel(const float* __restrict__ g3,
                                 const float* __restrict__ acc3,
                                 const float* __restrict__ dinv,
                                 const float* __restrict__ b3,
                                 float* __restrict__ out, int n)
{
    int i = blockIdx.x * blockDim.x + threadIdx.x;
    if (i >= n) return;
    const float z = dinv[i] * (acc3[i] + g3[i]) + b3[0];
    out[i] = 1.0f / (1.0f + expf(-z));
}

// ---------------- launcher ----------------
extern "C" void kernel_launch(void* const* d_in, const int* in_sizes, int n_in,
                              void* d_out, int out_size, void* d_ws, size_t ws_size,
                              hipStream_t stream)
{
    const float*     x_i = (const float*)d_in[0];
    const float*     x_j = (const float*)d_in[1];
    const long long* ei  = (const long long*)d_in[2];
    const float*     W1  = (const float*)d_in[3];
    const float*     b1  = (const float*)d_in[4];
    const float*     W2  = (const float*)d_in[5];
    const float*     b2  = (const float*)d_in[6];
    const float*     W3  = (const float*)d_in[7];
    const float*     b3  = (const float*)d_in[8];

    const int n = in_sizes[0] / 64;
    const int e = in_sizes[2] / 2;
    const long long* src = ei;
    const long long* dst = ei + e;

    // workspace layout (256-element aligned slabs)
    const size_t np = ((size_t)n + 255) & ~(size_t)255;
    float* ws   = (float*)d_ws;
    float* dinv = ws;                  // np
    float* g    = dinv + np;           // np*128
    float* acc  = g    + np * FEAT;    // np*128
    float* x2   = acc  + np * FEAT;    // np*128
    float* g3   = x2   + np * FEAT;    // np
    float* acc3 = g3   + np;           // np
    (void)ws_size; (void)n_in; (void)out_size;

    const int TB = 256;
    const long long nf4 = (long long)n * (FEAT / 4);
    const int gN   = (n + TB - 1) / TB;
    const int gE   = (e + TB - 1) / TB;
    const int gNF4 = (int)((nf4 + TB - 1) / TB);
    const int gN4  = (int)(((np / 4) + TB - 1) / TB);
    const int rowTiles = (n + 15) / 16;
    const int gWaveE = (e + 7) / 8;                 // 8 waves / block
    const int gWaveN = (n + 7) / 8;

    // normalization: dinv = rsqrt(1 + indegree)
    deg_init_kernel<<<gN, TB, 0, stream>>>(dinv, n);
    deg_edge_kernel<<<gE, TB, 0, stream>>>(dst, dinv, e);
    dinv_kernel<<<gN, TB, 0, stream>>>(dinv, n);

    // ---- layer 1 ----
    gemm_g_kernel<0><<<rowTiles, 256, 0, stream>>>(x_i, x_j, W1, dinv, g, n);
    zero4_kernel<<<gNF4, TB, 0, stream>>>((float4*)acc, nf4);
    scatter_kernel<<<gWaveE, TB, 0, stream>>>(g, src, dst, acc, e);
    finalize_kernel<<<gNF4, TB, 0, stream>>>((const float4*)g, (const float4*)acc,
                                             dinv, (const float4*)b1, (float4*)x2, n, 1);

    // ---- layer 2 ----
    gemm_g_kernel<1><<<rowTiles, 256, 0, stream>>>(x2, nullptr, W2, dinv, g, n);
    zero4_kernel<<<gNF4, TB, 0, stream>>>((float4*)acc, nf4);
    scatter_kernel<<<gWaveE, TB, 0, stream>>>(g, src, dst, acc, e);
    finalize_kernel<<<gNF4, TB, 0, stream>>>((const float4*)g, (const float4*)acc,
                                             dinv, (const float4*)b2, (float4*)x2, n, 1);

    // ---- layer 3 (128 -> 1) + sigmoid ----
    dot3_kernel<<<gWaveN, TB, 0, stream>>>(x2, W3, dinv, g3, n);
    zero4_kernel<<<gN4, TB, 0, stream>>>((float4*)acc3, (long long)(np / 4));
    scatter3_kernel<<<gE, TB, 0, stream>>>(g3, src, dst, acc3, e);
    finalize3_kernel<<<gN, TB, 0, stream>>>(g3, acc3, dinv, b3, (float*)d_out, n);
}